// VBRResidualVectorQuantize_56848187130002
// MI455X (gfx1250) — compile-verified
//
#include <hip/hip_runtime.h>
#include <hip/hip_bf16.h>
#include <math.h>

typedef float v2f __attribute__((ext_vector_type(2)));
typedef float v8f __attribute__((ext_vector_type(8)));

namespace {
constexpr int Bb = 16, Dd = 512, Tt = 4096;   // z: [B, D, T]
constexpr int Nn = 9, Kk = 1024, CD = 8;      // stages, codes, code dim
constexpr int FPB = 64;                       // frames (t) per block
constexpr int THREADS = 128;                  // 4 waves; each wave owns a 16-frame WMMA tile
constexpr int FS = FPB + 1;                   // residual row stride: resid[d][f]
constexpr float FEPS = 1e-12f;

constexpr long O_CODES = (long)Bb * Dd * Tt;                  // after z_q
constexpr long O_LAT   = O_CODES + (long)Bb * Nn * Tt;        // after codes
constexpr long O_LOSS  = O_LAT + (long)Bb * Nn * CD * Tt;     // two scalars

// LDS layout (float counts); all region starts are 16B aligned
constexpr int L_RESID = Dd * FS;       // 33280
constexpr int L_CBN   = Kk * CD;       // 8192  normalized codebook
constexpr int L_C2    = Kk;            // 1024  ||cb_n||^2
constexpr int L_WIN   = Dd * CD;       // 4096  W_in transposed [d][c]
constexpr int L_WOUT  = Dd * CD;       // 4096  W_out [d][c]
constexpr int L_BOUT  = Dd;            // 512
constexpr int L_BIN   = CD;            // 8
constexpr int L_ENCN  = FPB * CD;      // 512   normalized encodings
constexpr int L_E2    = FPB;           // 64    ||enc_n||^2
constexpr int L_Q     = FPB * CD;      // 512   selected (unnormalized) codes
constexpr int L_BIDX  = FPB;           // 64    argmin indices
constexpr int L_TOTAL = L_RESID + L_CBN + L_C2 + L_WIN + L_WOUT + L_BOUT +
                        L_BIN + L_ENCN + L_E2 + L_Q + L_BIDX;   // ~209 KB < 320 KB/WGP
} // namespace

__global__ __launch_bounds__(THREADS)
void rvq_fused_kernel(const float* __restrict__ z,
                      const float* __restrict__ W_in,
                      const float* __restrict__ b_in,
                      const float* __restrict__ W_out,
                      const float* __restrict__ b_out,
                      const float* __restrict__ cbk,
                      float* __restrict__ out,
                      float* __restrict__ wsPart)
{
    extern __shared__ float smem[];
    float* resid = smem;                     // [d][f], stride FS
    float* cbn   = resid + L_RESID;          // [k][c]
    float* c2    = cbn + L_CBN;              // [k]
    float* WinT  = c2 + L_C2;                // [d][c]
    float* Wout  = WinT + L_WIN;             // [d][c]
    float* boutS = Wout + L_WOUT;            // [d]
    float* binS  = boutS + L_BOUT;           // [c]
    float* encn  = binS + L_BIN;             // [f][c]
    float* e2s   = encn + L_ENCN;            // [f]
    float* qS    = e2s + L_E2;               // [f][c]
    int*   bidxS = (int*)(qS + L_Q);         // [f]

    const int tid  = threadIdx.x;
    const int lane = tid & 31;
    const int w    = tid >> 5;      // wave id: owns frames w*16 .. w*16+15
    const int fl   = lane & 15;     // frame-in-tile / B-column index
    const int g    = lane >> 4;     // half select (WMMA lane group)
    const int cb0  = g * 4;         // this lane's 4 channels in the 8-dim code space
    const int floc = w * 16 + fl;   // frame within block

    const long fg0 = (long)blockIdx.x * FPB;     // 64 consecutive t, same b
    const int  bI  = (int)(fg0 / Tt);
    const int  t0  = (int)(fg0 % Tt);
    const float* zb = z + (long)bI * Dd * Tt;

    // ---- load residual tile: resid[d][f] = z[b][d][t0+f] (coalesced over f)
    for (int idx = tid; idx < FPB * Dd; idx += THREADS) {
        const int f = idx & (FPB - 1);
        const int d = idx >> 6;
        resid[d * FS + f] = zb[(long)d * Tt + t0 + f];
    }
    __syncthreads();

    float lossAcc = 0.0f;

    for (int s = 0; s < Nn; ++s) {
        const float* Wi = W_in  + (long)s * CD * Dd;
        const float* Wo = W_out + (long)s * Dd * CD;
        const float* cb = cbk   + (long)s * Kk * CD;

        // ---- stage weights into LDS
        for (int x = tid; x < CD * Dd; x += THREADS) {
            const int c = x >> 9, d = x & (Dd - 1);
            WinT[d * CD + c] = Wi[x];    // transpose W_in -> [d][c]
            Wout[x] = Wo[x];             // W_out already [d][c]
        }
        for (int x = tid; x < Dd; x += THREADS) boutS[x] = b_out[(long)s * Dd + x];
        if (tid < CD) binS[tid] = b_in[s * CD + tid];
        // normalized codebook + squared norms
        for (int r = tid; r < Kk; r += THREADS) {
            float v[CD]; float ss = 0.f;
            #pragma unroll
            for (int c = 0; c < CD; ++c) { v[c] = cb[r * CD + c]; ss += v[c] * v[c]; }
            const float inv = 1.0f / fmaxf(sqrtf(ss), FEPS);
            float cc = 0.f;
            #pragma unroll
            for (int c = 0; c < CD; ++c) { const float t = v[c] * inv; cbn[r * CD + c] = t; cc += t * t; }
            c2[r] = cc;
        }
        // warm L2 for the next stage's weights while this stage computes
        if (s + 1 < Nn) {
            __builtin_prefetch(W_in  + (long)(s + 1) * CD * Dd + lane * 16, 0, 1);
            __builtin_prefetch(W_out + (long)(s + 1) * Dd * CD + lane * 16, 0, 1);
            __builtin_prefetch(cbk   + (long)(s + 1) * Kk * CD + tid * 64, 0, 1);
        }
        __syncthreads();

        // ---- in_proj: z_e[floc][cb0..cb0+3] = b_in + W_in . resid[:,floc]
        float ze[4] = { binS[cb0 + 0], binS[cb0 + 1], binS[cb0 + 2], binS[cb0 + 3] };
        for (int d = 0; d < Dd; ++d) {
            const float rv = resid[d * FS + floc];
            const float4 wv = *(const float4*)&WinT[d * CD + cb0];
            ze[0] = fmaf(wv.x, rv, ze[0]);
            ze[1] = fmaf(wv.y, rv, ze[1]);
            ze[2] = fmaf(wv.z, rv, ze[2]);
            ze[3] = fmaf(wv.w, rv, ze[3]);
        }

        // ---- L2 normalize (eps-clipped like reference)
        float ssum = ze[0]*ze[0] + ze[1]*ze[1] + ze[2]*ze[2] + ze[3]*ze[3];
        ssum += __shfl_xor(ssum, 16, 32);     // combine the two channel halves
        const float inv = 1.0f / fmaxf(sqrtf(ssum), FEPS);
        const float e2v = ssum * inv * inv;   // ||enc_n||^2
        #pragma unroll
        for (int j = 0; j < 4; ++j) encn[floc * CD + cb0 + j] = ze[j] * inv;
        if (g == 0) e2s[floc] = e2v;
        __syncthreads();

        // ---- dist + argmin via V_WMMA_F32_16X16X4_F32: [16 frames x 8] x [8 x 16 codes]
        // A 16x4 frag: lane<16 holds K={0,1}, lane>=16 K={2,3} (v2f)
        const v2f a0 = *(const v2f*)&encn[floc * CD + 2 * g];       // K-step 0..3
        const v2f a1 = *(const v2f*)&encn[floc * CD + 4 + 2 * g];   // K-step 4..7
        float e2r[8];
        #pragma unroll
        for (int v = 0; v < 8; ++v) e2r[v] = e2s[w * 16 + g * 8 + v];  // row M = v + 8*g

        float best[8]; int bidx[8];
        #pragma unroll
        for (int v = 0; v < 8; ++v) { best[v] = 3.4e38f; bidx[v] = 0x7fffffff; }

        // software-pipelined over 16-code chunks: prefetch next B frags + c2
        v2f  b0 = *(const v2f*)&cbn[fl * CD + 2 * g];
        v2f  b1 = *(const v2f*)&cbn[fl * CD + 4 + 2 * g];
        float cc = c2[fl];
        #pragma unroll 2
        for (int n0 = 0; n0 < Kk; n0 += 16) {
            const int np = ((n0 + 16) & (Kk - 1)) + fl;   // next chunk (wraps on last)
            const v2f  pb0 = *(const v2f*)&cbn[np * CD + 2 * g];
            const v2f  pb1 = *(const v2f*)&cbn[np * CD + 4 + 2 * g];
            const float pcc = c2[np];

            v8f acc = {0.f, 0.f, 0.f, 0.f, 0.f, 0.f, 0.f, 0.f};
            acc = __builtin_amdgcn_wmma_f32_16x16x4_f32(false, a0, false, b0, (short)0, acc, false, false);
            acc = __builtin_amdgcn_wmma_f32_16x16x4_f32(false, a1, false, b1, (short)0, acc, false, false);

            const int n = n0 + fl;  // this lane's code column
            #pragma unroll
            for (int v = 0; v < 8; ++v) {
                const float dist = e2r[v] - 2.0f * acc[v] + cc;     // same formula as reference
                if (dist < best[v]) { best[v] = dist; bidx[v] = n; } // strict < keeps first min
            }
            b0 = pb0; b1 = pb1; cc = pcc;
        }
        // cross-lane argmin over the 16 code columns (stay within 16-lane halves)
        #pragma unroll
        for (int m = 1; m <= 8; m <<= 1) {
            #pragma unroll
            for (int v = 0; v < 8; ++v) {
                const float ob = __shfl_xor(best[v], m, 32);
                const int   oi = __shfl_xor(bidx[v], m, 32);
                if (ob < best[v] || (ob == best[v] && oi < bidx[v])) { best[v] = ob; bidx[v] = oi; }
            }
        }
        if (fl == 0) {
            #pragma unroll
            for (int v = 0; v < 8; ++v) {
                const int fb = w * 16 + g * 8 + v;
                bidxS[fb] = bidx[v];
                out[O_CODES + (long)bI * Nn * Tt + (long)s * Tt + t0 + fb] = (float)bidx[v];
            }
        }
        __syncthreads();

        // ---- fetch selected (unnormalized) code, loss, latents
        {
            const int myidx = bidxS[floc];
            const float* qrow = cb + (long)myidx * CD + cb0;
            #pragma unroll
            for (int j = 0; j < 4; ++j) {
                const float qv = qrow[j];
                qS[floc * CD + cb0 + j] = qv;
                const float df = ze[j] - qv;
                lossAcc = fmaf(df, df, lossAcc);   // commitment == codebook in fwd
            }
            const long lb = O_LAT + (long)bI * Nn * CD * Tt + ((long)s * CD + cb0) * Tt + (t0 + floc);
            #pragma unroll
            for (int j = 0; j < 4; ++j) out[lb + (long)j * Tt] = ze[j];
        }
        __syncthreads();

        // ---- out_proj + residual update: lane owns f=floc, d in [g*256, g*256+256)
        {
            const float4 qa = *(const float4*)&qS[floc * CD];
            const float4 qb = *(const float4*)&qS[floc * CD + 4];
            const int d0 = g * (Dd / 2);
            for (int d = d0; d < d0 + Dd / 2; ++d) {
                const float4 wa = *(const float4*)&Wout[d * CD];
                const float4 wb = *(const float4*)&Wout[d * CD + 4];
                float o = boutS[d];
                o = fmaf(wa.x, qa.x, o); o = fmaf(wa.y, qa.y, o);
                o = fmaf(wa.z, qa.z, o); o = fmaf(wa.w, qa.w, o);
                o = fmaf(wb.x, qb.x, o); o = fmaf(wb.y, qb.y, o);
                o = fmaf(wb.z, qb.z, o); o = fmaf(wb.w, qb.w, o);
                resid[d * FS + floc] -= o;
            }
        }
        __syncthreads();
    }

    // ---- z_q = z - residual_final (coalesced over f)
    for (int idx = tid; idx < FPB * Dd; idx += THREADS) {
        const int f = idx & (FPB - 1);
        const int d = idx >> 6;
        out[(long)bI * Dd * Tt + (long)d * Tt + t0 + f] =
            zb[(long)d * Tt + t0 + f] - resid[d * FS + f];
    }

    // ---- deterministic per-block loss partial -> d_ws
    #pragma unroll
    for (int m = 1; m < 32; m <<= 1) lossAcc += __shfl_xor(lossAcc, m, 32);
    __shared__ float lred[THREADS / 32];
    if (lane == 0) lred[w] = lossAcc;
    __syncthreads();
    if (tid == 0) {
        float a = 0.f;
        #pragma unroll
        for (int i = 0; i < THREADS / 32; ++i) a += lred[i];
        wsPart[blockIdx.x] = a;
    }
}

__global__ __launch_bounds__(256)
void rvq_loss_reduce_kernel(const float* __restrict__ part, float* __restrict__ out)
{
    __shared__ float sm[256];
    float a = 0.f;
    for (int i = threadIdx.x; i < (Bb * Tt) / FPB; i += 256) a += part[i];
    sm[threadIdx.x] = a;
    __syncthreads();
    for (int s2 = 128; s2 > 0; s2 >>= 1) {
        if ((int)threadIdx.x < s2) sm[threadIdx.x] += sm[threadIdx.x + s2];
        __syncthreads();
    }
    if (threadIdx.x == 0) {
        const float v = sm[0] / (float)((long)Bb * CD * Tt);  // sum_stages(mean) == total/(B*CD*T)
        out[O_LOSS]     = v;   // commitment_loss
        out[O_LOSS + 1] = v;   // codebook_loss (identical forward value)
    }
}

extern "C" void kernel_launch(void* const* d_in, const int* in_sizes, int n_in,
                              void* d_out, int out_size, void* d_ws, size_t ws_size,
                              hipStream_t stream)
{
    (void)in_sizes; (void)n_in; (void)out_size; (void)ws_size;
    const float* z     = (const float*)d_in[0];
    const float* W_in  = (const float*)d_in[1];
    const float* b_in  = (const float*)d_in[2];
    const float* W_out = (const float*)d_in[3];
    const float* b_out = (const float*)d_in[4];
    const float* cbk   = (const float*)d_in[5];
    float* out  = (float*)d_out;
    float* part = (float*)d_ws;   // 1024 floats of scratch

    const int nblocks = (Bb * Tt) / FPB;                 // 1024 blocks of 64 frames
    const size_t smemBytes = (size_t)L_TOTAL * sizeof(float);
    rvq_fused_kernel<<<nblocks, THREADS, smemBytes, stream>>>(
        z, W_in, b_in, W_out, b_out, cbk, out, part);
    rvq_loss_reduce_kernel<<<1, 256, 0, stream>>>(part, out);
}